// GAT_35510789603475
// MI455X (gfx1250) — compile-verified
//
#include <hip/hip_runtime.h>
#include <hip/hip_bf16.h>

// ---------------------------------------------------------------------------
// GAT (2x GATConv + linear) for MI455X / gfx1250.
// Dense GEMMs use V_WMMA_F32_16X16X4_F32 (fp32 WMMA, full precision; the
// GEMMs are ~2.6 GFLOP total, far below any compute ceiling). Edge passes are
// memory-bound (~1 GB traffic -> ~45us @ 23.3 TB/s) and use float atomics
// matching the reference's segment_max / segment_sum semantics.
//
// Round-1: compile-time RAGGED guard -> no exec-mask churn in the WMMA loop.
// Round-2: K and M are now template constants and the K loop fully unrolls,
// so A/B addresses fold into immediate offsets off two base pointers and the
// per-iteration 64-bit address math disappears from the hot loop.
// ---------------------------------------------------------------------------

#define NN   50000      // nodes
#define FF   64         // in features
#define EE   800000     // edges
#define HH   2          // heads
#define CC   64         // per-head channels
#define DD   128        // HH*CC
#define OUTF 10
#define NEG_SLOPE 0.2f

typedef float v2f __attribute__((ext_vector_type(2)));
typedef float v8f __attribute__((ext_vector_type(8)));

// ---------------------------------------------------------------------------
// WMMA fp32 GEMM: C[N,M] = A[N,K] @ B[K,M] (+bias). One wave per 16x16 tile.
// A-frag (16x4 f32): lane<16 -> row=lane, {K,K+1}; lane>=16 -> row=lane-16,
// {K+2,K+3}. B-frag (4x16): lane&15 = col, vgpr pair holds the K rows with
// hi-half lanes holding K+2/K+3. C (16x16 f32): vgpr r -> row r + 8*(lane>=16),
// col = lane&15. N must be a multiple of 16 (50000 = 3125*16).
// RAGGED=false: M multiple of 16, no guards in the hot loop.
// RAGGED=true (final M=10): branchless clamp+mask on B columns, guarded store.
// ---------------------------------------------------------------------------
template <int K, int M, bool RAGGED>
__global__ void gat_wmma_gemm_f32(const float* __restrict__ A,
                                  const float* __restrict__ B,
                                  float* __restrict__ C,
                                  const float* __restrict__ bias) {
  const int lane = threadIdx.x;
  const int lo = lane & 15;
  const int hi = lane >> 4;
  const int r0 = blockIdx.x << 4;
  const int c0 = blockIdx.y << 4;

  const int  arow  = r0 + lo;
  const int  bcol  = c0 + lo;
  const bool cok   = !RAGGED || (bcol < M);
  const int  bsafe = RAGGED ? (cok ? bcol : 0) : bcol;   // always in-bounds
  const float bmask = (RAGGED && !cok) ? 0.0f : 1.0f;

  const float* Ap = A + (long long)arow * K + (hi << 1); // lane's A base
  const float* Bp = B + (hi << 1) * M + bsafe;           // lane's B base

  v8f acc = {};
#pragma unroll
  for (int k0 = 0; k0 < K; k0 += 4) {
    v2f a, b;
    a.x = Ap[k0];
    a.y = Ap[k0 + 1];
    b.x = Bp[k0 * M];
    b.y = Bp[(k0 + 1) * M];
    if (RAGGED) { b.x *= bmask; b.y *= bmask; }
    acc = __builtin_amdgcn_wmma_f32_16x16x4_f32(
        /*neg_a=*/false, a, /*neg_b=*/false, b,
        /*c_mod=*/(short)0, acc, /*reuse_a=*/false, /*reuse_b=*/false);
  }

  if (cok) {
    const float bv = bias ? bias[bsafe] : 0.0f;
    float* Cp = C + (long long)(r0 + (hi << 3)) * M + bcol;
#pragma unroll
    for (int r = 0; r < 8; ++r) {
      Cp[r * M] = acc[r] + bv;
    }
  }
}

// alpha_src[n,h] = <z[n,h,:], a_src[h,:]>, alpha_dst likewise.
__global__ void gat_attn_coeff(const float* __restrict__ z,
                               const float* __restrict__ a_src,
                               const float* __restrict__ a_dst,
                               float* __restrict__ alpha_src,
                               float* __restrict__ alpha_dst) {
  const int t = blockIdx.x * blockDim.x + threadIdx.x;
  if (t >= NN * HH) return;
  const int n = t >> 1;
  const int h = t & 1;
  const float* zp = z + (long long)n * DD + h * CC;
  const float* as = a_src + h * CC;
  const float* ad = a_dst + h * CC;
  float s0 = 0.0f, s1 = 0.0f;
#pragma unroll 8
  for (int c = 0; c < CC; ++c) {
    const float zv = zp[c];
    s0 += zv * as[c];
    s1 += zv * ad[c];
  }
  alpha_src[t] = s0;
  alpha_dst[t] = s1;
}

__device__ __forceinline__ float leaky(float v) {
  return v > 0.0f ? v : NEG_SLOPE * v;
}
// Order-preserving float->uint key (atomicMax-able; memset(0) == -inf).
__device__ __forceinline__ unsigned fkey(float f) {
  const unsigned u = __float_as_uint(f);
  return (u & 0x80000000u) ? ~u : (u | 0x80000000u);
}
__device__ __forceinline__ float funkey(unsigned k) {
  return __uint_as_float((k & 0x80000000u) ? (k & 0x7FFFFFFFu) : ~k);
}

// Pass 1: segment max of leaky_relu(alpha_src[src]+alpha_dst[dst]) over dst.
__global__ void gat_edge_max(const int* __restrict__ src,
                             const int* __restrict__ dst,
                             const float* __restrict__ asrc,
                             const float* __restrict__ adst,
                             unsigned* __restrict__ menc) {
  const int t = blockIdx.x * blockDim.x + threadIdx.x;
  if (t >= EE * HH) return;
  const int e = t >> 1;
  const int h = t & 1;
  const int s = src[e];
  const int d = dst[e];
  const float v = leaky(asrc[s * HH + h] + adst[d * HH + h]);
  atomicMax(&menc[d * HH + h], fkey(v));
}

// Pass 2: ex = exp(e - m[dst]); cache per-edge; segment-sum into denom.
__global__ void gat_edge_expsum(const int* __restrict__ src,
                                const int* __restrict__ dst,
                                const float* __restrict__ asrc,
                                const float* __restrict__ adst,
                                const unsigned* __restrict__ menc,
                                float* __restrict__ exbuf,
                                float* __restrict__ denom) {
  const int t = blockIdx.x * blockDim.x + threadIdx.x;
  if (t >= EE * HH) return;
  const int e = t >> 1;
  const int h = t & 1;
  const int s = src[e];
  const int d = dst[e];
  const float v = leaky(asrc[s * HH + h] + adst[d * HH + h]);
  const float m = funkey(menc[d * HH + h]);
  const float ex = __expf(v - m);
  exbuf[t] = ex;
  atomicAdd(&denom[d * HH + h], ex);
}

// Pass 3: acc[dst, j] += z[src, j] * ex/(denom+1e-16). Thread per (edge, j).
__global__ void gat_edge_aggregate(const int* __restrict__ src,
                                   const int* __restrict__ dst,
                                   const float* __restrict__ z,
                                   const float* __restrict__ exbuf,
                                   const float* __restrict__ denom,
                                   float* __restrict__ acc) {
  const int t = blockIdx.x * blockDim.x + threadIdx.x;  // EE*DD = 102.4M < 2^31
  if (t >= EE * DD) return;
  const int e = t >> 7;
  const int j = t & (DD - 1);
  const int h = j >> 6;
  const int s = src[e];
  const int d = dst[e];
  const float alpha = exbuf[e * HH + h] / (denom[d * HH + h] + 1e-16f);
  atomicAdd(&acc[(long long)d * DD + j], z[(long long)s * DD + j] * alpha);
}

// out = acc + bias (optional relu), in place.
__global__ void gat_bias_act(float* __restrict__ buf,
                             const float* __restrict__ bias, int do_relu) {
  const int t = blockIdx.x * blockDim.x + threadIdx.x;
  if (t >= NN * DD) return;
  float v = buf[t] + bias[t & (DD - 1)];
  if (do_relu) v = fmaxf(v, 0.0f);
  buf[t] = v;
}

extern "C" void kernel_launch(void* const* d_in, const int* in_sizes, int n_in,
                              void* d_out, int out_size, void* d_ws, size_t ws_size,
                              hipStream_t stream) {
  (void)in_sizes; (void)n_in; (void)out_size; (void)ws_size;

  const float* x    = (const float*)d_in[0];
  const int*   eidx = (const int*)d_in[1];        // [2, E]
  const int*   src  = eidx;
  const int*   dst  = eidx + EE;
  // d_in[2] = edge_attr (ignored, edge_dim=None)
  const float* W1   = (const float*)d_in[3];
  const float* as1  = (const float*)d_in[4];
  const float* ad1  = (const float*)d_in[5];
  const float* b1   = (const float*)d_in[6];
  const float* W2   = (const float*)d_in[7];
  const float* as2  = (const float*)d_in[8];
  const float* ad2  = (const float*)d_in[9];
  const float* b2   = (const float*)d_in[10];
  const float* Wl   = (const float*)d_in[11];
  const float* bl   = (const float*)d_in[12];
  float* out = (float*)d_out;

  // Workspace carve-up (~60 MB).
  float*    bufA  = (float*)d_ws;                       // z of current layer
  float*    bufB  = bufA + (size_t)NN * DD;             // acc -> h of layer
  float*    aSrc  = bufB + (size_t)NN * DD;             // [N,H]
  float*    aDst  = aSrc + (size_t)NN * HH;             // [N,H]
  unsigned* menc  = (unsigned*)(aDst + (size_t)NN * HH);// [N,H] encoded max
  float*    denom = (float*)(menc + (size_t)NN * HH);   // [N,H]
  float*    exbuf = denom + (size_t)NN * HH;            // [E,H]

  const int ROWT = NN / 16;  // 3125 row tiles
  const dim3 wave(32);
  const int TPB = 256;
  const int nbNH = (NN * HH + TPB - 1) / TPB;
  const int nbEH = (EE * HH + TPB - 1) / TPB;
  const int nbED = (EE * DD) / TPB;      // 400000, exact
  const int nbND = (NN * DD) / TPB;      // 25000, exact

  // ---------------- layer 1 ----------------
  hipMemsetAsync(bufB,  0, (size_t)NN * DD * sizeof(float), stream);
  hipMemsetAsync(menc,  0, (size_t)NN * HH * sizeof(unsigned), stream); // -inf
  hipMemsetAsync(denom, 0, (size_t)NN * HH * sizeof(float), stream);

  gat_wmma_gemm_f32<FF, DD, false>
      <<<dim3(ROWT, DD / 16), wave, 0, stream>>>(x, W1, bufA, nullptr);
  gat_attn_coeff<<<nbNH, TPB, 0, stream>>>(bufA, as1, ad1, aSrc, aDst);
  gat_edge_max<<<nbEH, TPB, 0, stream>>>(src, dst, aSrc, aDst, menc);
  gat_edge_expsum<<<nbEH, TPB, 0, stream>>>(src, dst, aSrc, aDst, menc, exbuf, denom);
  gat_edge_aggregate<<<nbED, TPB, 0, stream>>>(src, dst, bufA, exbuf, denom, bufB);
  gat_bias_act<<<nbND, TPB, 0, stream>>>(bufB, b1, /*relu=*/1);   // bufB = h1

  // ---------------- layer 2 ----------------
  gat_wmma_gemm_f32<DD, DD, false>
      <<<dim3(ROWT, DD / 16), wave, 0, stream>>>(bufB, W2, bufA, nullptr);
  hipMemsetAsync(bufB,  0, (size_t)NN * DD * sizeof(float), stream);  // acc2
  hipMemsetAsync(menc,  0, (size_t)NN * HH * sizeof(unsigned), stream);
  hipMemsetAsync(denom, 0, (size_t)NN * HH * sizeof(float), stream);

  gat_attn_coeff<<<nbNH, TPB, 0, stream>>>(bufA, as2, ad2, aSrc, aDst);
  gat_edge_max<<<nbEH, TPB, 0, stream>>>(src, dst, aSrc, aDst, menc);
  gat_edge_expsum<<<nbEH, TPB, 0, stream>>>(src, dst, aSrc, aDst, menc, exbuf, denom);
  gat_edge_aggregate<<<nbED, TPB, 0, stream>>>(src, dst, bufA, exbuf, denom, bufB);
  gat_bias_act<<<nbND, TPB, 0, stream>>>(bufB, b2, /*relu=*/0);   // bufB = h2

  // ---------------- final linear: out = h2 @ Wl + bl ----------------
  gat_wmma_gemm_f32<DD, OUTF, true>
      <<<dim3(ROWT, 1), wave, 0, stream>>>(bufB, Wl, out, bl);
}